// MAGNet_3582002725419
// MI455X (gfx1250) — compile-verified
//
#include <hip/hip_runtime.h>
#include <hip/hip_bf16.h>

typedef __attribute__((ext_vector_type(2))) float v2f;
typedef __attribute__((ext_vector_type(8))) float v8f;

#define MASKV (-9e15f)

__device__ __forceinline__ v8f wmma_f32_k4(v2f a, v2f b, v8f c) {
  // D = A(16x4 f32) * B(4x16 f32) + C(16x16 f32)
  return __builtin_amdgcn_wmma_f32_16x16x4_f32(false, a, false, b, (short)0, c,
                                               false, false);
}

// out[N,H] = ACT( [X1|X2][N,K1+K2] @ W[K,H] + bias[H] ), H = 128.
// grid.x = N/16, block = 256 (8 wave32); wave w owns output cols [16w,16w+16).
// K1,K2 compile-time -> fully unrolled, all loads at immediate offsets.
template <int K1, int K2, int ACT>
__global__ void gemm_bias_act(const float* __restrict__ X1,
                              const float* __restrict__ X2,
                              const float* __restrict__ W,
                              const float* __restrict__ bias,
                              float* __restrict__ out) {
  constexpr int H = 128;
  int tid  = threadIdx.x;
  int lane = tid & 31;
  int wv   = tid >> 5;
  int rbase = blockIdx.x * 16;
  int cbase = wv * 16;
  int n16  = lane & 15;
  int klo  = (lane < 16) ? 0 : 2;

  float bval = bias[cbase + n16];
  v8f c;
#pragma unroll
  for (int r = 0; r < 8; ++r) c[r] = bval;

  // A: lane holds row (rbase+n16), K-pair starting at klo (+k0)
  const float* pA  = X1 + (size_t)(rbase + n16) * K1 + klo;
  // B: lane holds col (cbase+n16), rows klo,klo+1 (+k0)
  const float* pB0 = W + (size_t)klo * H + cbase + n16;
  const float* pB1 = pB0 + H;
#pragma unroll
  for (int k0 = 0; k0 < K1; k0 += 4) {
    v2f a = *(const v2f*)(pA + k0);
    v2f b;
    b.x = pB0[(size_t)k0 * H];
    b.y = pB1[(size_t)k0 * H];
    c = wmma_f32_k4(a, b, c);
  }
  if (K2 > 0) {  // concat tail [h|z]: separate unrolled loop, no per-step branch
    const float* pA2  = X2 + (size_t)(rbase + n16) * (K2 ? K2 : 1) + klo;
    const float* pB0b = W + (size_t)(K1 + klo) * H + cbase + n16;
    const float* pB1b = pB0b + H;
#pragma unroll
    for (int k0 = 0; k0 < K2; k0 += 4) {
      v2f a = *(const v2f*)(pA2 + k0);
      v2f b;
      b.x = pB0b[(size_t)k0 * H];
      b.y = pB1b[(size_t)k0 * H];
      c = wmma_f32_k4(a, b, c);
    }
  }

  int rowoff = (lane < 16) ? 0 : 8;
#pragma unroll
  for (int r = 0; r < 8; ++r) {
    float v = c[r];
    if (ACT == 1) v = v > 0.f ? v : 0.f;              // relu
    if (ACT == 2) v = 1.f / (1.f + __expf(-v));       // sigmoid
    out[(size_t)(rbase + r + rowoff) * H + cbase + n16] = v;
  }
}

// gm = gate * raw ; lossAcc += sum(gate)
__global__ void gate_mul_loss(const float* __restrict__ gate,
                              const float* __restrict__ raw,
                              float* __restrict__ gm, float* lossAcc, int total) {
  __shared__ float red[256];
  int idx = blockIdx.x * 256 + threadIdx.x;
  float g = 0.f;
  if (idx < total) { g = gate[idx]; gm[idx] = g * raw[idx]; }
  red[threadIdx.x] = g;
  __syncthreads();
  for (int off = 128; off > 0; off >>= 1) {
    if (threadIdx.x < off) red[threadIdx.x] += red[threadIdx.x + off];
    __syncthreads();
  }
  if (threadIdx.x == 0) atomicAdd(lossAcc, red[0]);
}

// Fold attention vectors: wk = Wk@a1, ck = bk.a1 ; wq = Wq@a2, cq = bq.a2
__global__ void fold_attn_vec(const float* __restrict__ Wk, const float* __restrict__ bk,
                              const float* __restrict__ a1,
                              const float* __restrict__ Wq, const float* __restrict__ bq,
                              const float* __restrict__ a2,
                              float* wk, float* ck, float* wq, float* cq) {
  int t = threadIdx.x;  // 256
  if (t < 128) {
    float s = 0.f;
    for (int hh = 0; hh < 128; ++hh) s += Wk[t * 128 + hh] * a1[hh];
    wk[t] = s;
    if (t == 0) {
      float cc = 0.f;
      for (int hh = 0; hh < 128; ++hh) cc += bk[hh] * a1[hh];
      *ck = cc;
    }
  } else {
    int r = t - 128;
    float s = 0.f;
    for (int hh = 0; hh < 128; ++hh) s += Wq[r * 128 + hh] * a2[hh];
    wq[r] = s;
    if (r == 0) {
      float cc = 0.f;
      for (int hh = 0; hh < 128; ++hh) cc += bq[hh] * a2[hh];
      *cq = cc;
    }
  }
}

// kalpha[row] = h[row,:].wk + ck ; qbeta[row] = gm[row,:].wq + cq   (block=128, grid=N)
__global__ void dual_gemv(const float* __restrict__ h, const float* __restrict__ gm,
                          const float* __restrict__ wk, const float* __restrict__ ck,
                          const float* __restrict__ wq, const float* __restrict__ cq,
                          float* kalpha, float* qbeta) {
  constexpr int H = 128;
  __shared__ float ra[128], rb[128];
  int row = blockIdx.x, t = threadIdx.x;
  ra[t] = h[(size_t)row * H + t] * wk[t];
  rb[t] = gm[(size_t)row * H + t] * wq[t];
  __syncthreads();
  for (int off = 64; off > 0; off >>= 1) {
    if (t < off) { ra[t] += ra[t + off]; rb[t] += rb[t + off]; }
    __syncthreads();
  }
  if (t == 0) { kalpha[row] = ra[0] + *ck; qbeta[row] = rb[0] + *cq; }
}

// Online softmax row stats over j: mrow[i] = max_j e(i,j), sinv[i] = 1/sum_j exp(e-m)
__global__ void row_stats(const int* __restrict__ adj, const float* __restrict__ kalpha,
                          const float* __restrict__ qbeta,
                          float* mrow, float* sinv) {
  constexpr int N = 8192;
  __shared__ float sm[256], ss[256];
  int i = blockIdx.x;
  int tid = threadIdx.x;
  float qb = qbeta[i];
  const int* ap = adj + (size_t)i * N + tid;  // NT stream: adj never fits L2
  float mloc = -3.0e38f, sloc = 0.f;
  for (int j = tid; j < N; j += 256, ap += 256) {
    int a = __builtin_nontemporal_load(ap);
    float t = kalpha[j] + qb;
    float e = t > 0.f ? t : 0.01f * t;
    e = (a > 0) ? e : MASKV;
    if (e > mloc) { sloc = sloc * __expf(mloc - e) + 1.f; mloc = e; }
    else sloc += __expf(e - mloc);
  }
  sm[tid] = mloc; ss[tid] = sloc;
  __syncthreads();
  for (int off = 128; off > 0; off >>= 1) {
    if (tid < off) {
      float m1 = sm[tid], s1 = ss[tid];
      float m2 = sm[tid + off], s2 = ss[tid + off];
      float mn = fmaxf(m1, m2);
      sm[tid] = mn;
      ss[tid] = s1 * __expf(m1 - mn) + s2 * __expf(m2 - mn);
    }
    __syncthreads();
  }
  if (tid == 0) { mrow[i] = sm[0]; sinv[i] = 1.f / ss[0]; }
}

// agg = attn.T @ gm via WMMA, fused residual: h = (agg + h) * 0.5
// grid.x = N/16 (16 output j-rows per block), block = 256 (8 waves x 16 hid cols).
// 16x64 weight tile computed cooperatively into LDS (each exp computed once),
// then all 8 waves feed WMMA from LDS (stride-66 rows -> aligned ds_load_b64).
__global__ void attn_agg(const int* __restrict__ adj, const float* __restrict__ kalpha,
                         const float* __restrict__ qbeta, const float* __restrict__ mrow,
                         const float* __restrict__ sinv, const float* __restrict__ gm,
                         float* __restrict__ h) {
  constexpr int N = 8192, H = 128;
  __shared__ __align__(16) float Wt[16 * 66];
  int tid  = threadIdx.x;
  int lane = tid & 31;
  int wv   = tid >> 5;
  int jbase = blockIdx.x * 16;
  int cbase = wv * 16;
  int n16  = lane & 15;
  int klo  = (lane < 16) ? 0 : 2;

  int jloc  = tid & 15;   // weight-compute: fixed j per thread
  int iisub = tid >> 4;   // 0..15
  float kaj = kalpha[jbase + jloc];
  float* wslot = &Wt[jloc * 66 + iisub];

  // loop-carried base pointers: strides become immediates, no per-iter mul
  const int*   adjp = adj + (size_t)iisub * N + jbase + jloc;
  const float* qbp  = qbeta + iisub;
  const float* mrp  = mrow + iisub;
  const float* svp  = sinv + iisub;
  const float* gmp  = gm + (size_t)klo * H + cbase + n16;

  v8f c = {};
  for (int i0 = 0; i0 < N; i0 += 64) {
    // 16x64 weight tile: w(i,j) = exp(e(i,j) - m_i) / s_i
#pragma unroll
    for (int r = 0; r < 4; ++r) {
      int a = __builtin_nontemporal_load(adjp + (size_t)(r * 16) * N);
      float t = kaj + qbp[r * 16];
      float e = t > 0.f ? t : 0.01f * t;
      e = (a > 0) ? e : MASKV;
      wslot[r * 16] = __expf(e - mrp[r * 16]) * svp[r * 16];
    }
    __syncthreads();
#pragma unroll
    for (int ik = 0; ik < 64; ik += 4) {
      v2f a = *(const v2f*)&Wt[n16 * 66 + ik + klo];
      v2f b;
      b.x = gmp[(size_t)ik * H];
      b.y = gmp[(size_t)ik * H + H];
      c = wmma_f32_k4(a, b, c);
    }
    __syncthreads();
    adjp += (size_t)64 * N;
    qbp += 64; mrp += 64; svp += 64;
    gmp += 64 * H;
  }

  int rowoff = (lane < 16) ? 0 : 8;
#pragma unroll
  for (int r = 0; r < 8; ++r) {
    size_t idx = (size_t)(jbase + r + rowoff) * H + cbase + n16;
    h[idx] = (c[r] + h[idx]) * 0.5f;  // each element owned by exactly one block
  }
}

__global__ void init_loss(float* lossAcc) {
  if (threadIdx.x == 0 && blockIdx.x == 0) *lossAcc = 0.f;
}

__global__ void finalize(const float* __restrict__ h, const float* __restrict__ lossAcc,
                         float* __restrict__ out, int total, float scale) {
  int idx = blockIdx.x * 256 + threadIdx.x;
  if (idx < total) out[idx] = h[idx];
  if (idx == 0) out[total] = *lossAcc * scale;
}

extern "C" void kernel_launch(void* const* d_in, const int* in_sizes, int n_in,
                              void* d_out, int out_size, void* d_ws, size_t ws_size,
                              hipStream_t stream) {
  (void)in_sizes; (void)n_in; (void)out_size; (void)ws_size;
  constexpr int N = 8192, IN_DIM = 128, HID = 128, TASK = 64, L = 2;

  const float* h_in = (const float*)d_in[0];
  const int*   adj  = (const int*)d_in[1];
  const float* z    = (const float*)d_in[2];
  const float* Wp   = (const float*)d_in[3];
  const float* bp   = (const float*)d_in[4];
  const float* Wm   = (const float*)d_in[5];
  const float* bm   = (const float*)d_in[6];
  const float* Wg   = (const float*)d_in[7];
  const float* bg   = (const float*)d_in[8];
  const float* Wk   = (const float*)d_in[9];
  const float* bk   = (const float*)d_in[10];
  const float* Wq   = (const float*)d_in[11];
  const float* bq   = (const float*)d_in[12];
  const float* a1   = (const float*)d_in[13];
  const float* a2   = (const float*)d_in[14];

  float* ws = (float*)d_ws;
  size_t o = 0;
  float* hbuf   = ws + o; o += (size_t)N * HID;
  float* raw    = ws + o; o += (size_t)N * HID;
  float* gate   = ws + o; o += (size_t)N * HID;
  float* gm     = ws + o; o += (size_t)N * HID;
  float* kalpha = ws + o; o += N;
  float* qbeta  = ws + o; o += N;
  float* mrowv  = ws + o; o += N;
  float* sinv   = ws + o; o += N;
  float* wkv    = ws + o; o += 128;
  float* wqv    = ws + o; o += 128;
  float* ckv    = ws + o; o += 1;
  float* cqv    = ws + o; o += 1;
  float* loss   = ws + o; o += 1;

  init_loss<<<1, 64, 0, stream>>>(loss);

  // h = h_in @ Wp + bp
  gemm_bias_act<IN_DIM, 0, 0><<<N / 16, 256, 0, stream>>>(h_in, nullptr, Wp, bp, hbuf);

  for (int l = 0; l < L; ++l) {
    const float* Wml = Wm + (size_t)l * HID * HID;
    const float* bml = bm + (size_t)l * HID;
    const float* Wgl = Wg + (size_t)l * (HID + TASK) * HID;
    const float* bgl = bg + (size_t)l * HID;
    const float* Wkl = Wk + (size_t)l * HID * HID;
    const float* bkl = bk + (size_t)l * HID;
    const float* Wql = Wq + (size_t)l * HID * HID;
    const float* bql = bq + (size_t)l * HID;
    const float* a1l = a1 + (size_t)l * HID;
    const float* a2l = a2 + (size_t)l * HID;

    // raw = relu(h @ Wm + bm)
    gemm_bias_act<HID, 0, 1><<<N / 16, 256, 0, stream>>>(hbuf, nullptr, Wml, bml, raw);
    // gate = sigmoid([h|z] @ Wg + bg)
    gemm_bias_act<HID, TASK, 2><<<N / 16, 256, 0, stream>>>(hbuf, z, Wgl, bgl, gate);
    // gm = gate*raw ; loss += sum(gate)
    gate_mul_loss<<<(N * HID) / 256, 256, 0, stream>>>(gate, raw, gm, loss, N * HID);
    // fold: k@a1 == h@(Wk@a1)+bk.a1 ; q@a2 == gm@(Wq@a2)+bq.a2
    fold_attn_vec<<<1, 256, 0, stream>>>(Wkl, bkl, a1l, Wql, bql, a2l, wkv, ckv, wqv, cqv);
    dual_gemv<<<N, 128, 0, stream>>>(hbuf, gm, wkv, ckv, wqv, cqv, kalpha, qbeta);
    // pass 1 over adj: per-row softmax stats
    row_stats<<<N, 256, 0, stream>>>(adj, kalpha, qbeta, mrowv, sinv);
    // pass 2 over adj: agg = attn.T @ gm via WMMA, fused h = (agg+h)/2
    attn_agg<<<N / 16, 256, 0, stream>>>(adj, kalpha, qbeta, mrowv, sinv, gm, hbuf);
  }

  finalize<<<(N * HID + 255) / 256, 256, 0, stream>>>(
      hbuf, loss, (float*)d_out, N * HID, 1.0f / ((float)N * HID * L));
}